// CGIterator_49203145343000
// MI455X (gfx1250) — compile-verified
//
#include <hip/hip_runtime.h>

typedef float f32x2 __attribute__((ext_vector_type(2)));
typedef float f32x8 __attribute__((ext_vector_type(8)));

#if defined(__has_builtin)
#if __has_builtin(__builtin_amdgcn_global_load_async_to_lds_b128) && \
    __has_builtin(__builtin_amdgcn_s_wait_asynccnt)
#define HAVE_GFX1250_ASYNC 1
#endif
#endif
#ifndef HAVE_GFX1250_ASYNC
#define HAVE_GFX1250_ASYNC 0
#endif

// ---------------------------------------------------------------------------
// WMMA fp32 GEMM:  C[M,N] = A[M,K] * B[K,N]  (+ R[M,N] if R != nullptr)
// Block: 256 threads = 8 waves. Tile: 64(M) x 32(N), K-step 32.
// Wave w -> 16x16 sub-tile (row tile w>>1, col tile w&1).
// Async path: tiles copied global->LDS with GLOBAL_LOAD_ASYNC_TO_LDS_B128,
// double-buffered, so the copy of tile i+1 overlaps the 8 WMMAs of tile i.
// Fragment layout (CDNA5 ISA 7.12.2):
//   A 16x4 fp32: VGPR0 = K0(lanes0-15)/K2(lanes16-31), VGPR1 = K1/K3, M=lane%16
//   B 4x16 fp32: VGPR0 = rows K0/K2 across lane halves, VGPR1 = K1/K3, N=lane%16
// Requires K % 32 == 0, N % 32 == 0 (true for all GEMMs here); M arbitrary.
// ---------------------------------------------------------------------------
#define TILE_M 64
#define TILE_N 32
#define TILE_K 32
#define LDSS 36   // row stride (floats): 16B-aligned rows for async B128,
                  // and 36*i mod 64 distinct for i=0..15 -> conflict-free frags

#if HAVE_GFX1250_ASYNC
typedef int v4i __attribute__((vector_size(16)));
typedef __attribute__((address_space(1))) v4i* gptr_v4i;
typedef __attribute__((address_space(3))) v4i* lptr_v4i;

__device__ __forceinline__ void async_b128(const float* gsrc, float* ldst)
{
    __builtin_amdgcn_global_load_async_to_lds_b128(
        (gptr_v4i)gsrc, (lptr_v4i)ldst, 0, 0);
}
#endif

__global__ __launch_bounds__(256) void gemm_wmma_f32(
    const float* __restrict__ A, const float* __restrict__ B,
    const float* __restrict__ R, float* __restrict__ C,
    int M, int K, int N)
{
    const int tid  = threadIdx.x;
    const int lane = tid & 31;
    const int wave = tid >> 5;
    const int hl   = lane >> 4;       // lane half: 0/1
    const int l16  = lane & 15;
    const int wr   = wave >> 1;       // 0..3
    const int wc   = wave & 1;        // 0..1

    const int m0 = blockIdx.x * TILE_M;
    const int n0 = blockIdx.y * TILE_N;

    f32x8 acc = {};

    const int ar = wr * 16 + l16;     // fragment row within A tile
    const int bc = wc * 16 + l16;     // fragment col within B tile

#if HAVE_GFX1250_ASYNC
    __shared__ float As[2][TILE_M * LDSS];
    __shared__ float Bs[2][TILE_K * LDSS];

    // per-thread staging coordinates (fixed across K tiles)
    const int sr = tid >> 3;              // 0..31
    const int sc = (tid & 7) << 2;        // 0,4,...,28
    int gr0 = m0 + sr;      gr0 = (gr0 < M) ? gr0 : (M - 1);
    int gr1 = m0 + sr + 32; gr1 = (gr1 < M) ? gr1 : (M - 1);
    const float* pA0 = A + (long)gr0 * K + sc;
    const float* pA1 = A + (long)gr1 * K + sc;
    const float* pB  = B + (long)sr * N + n0 + sc;

    const int ntiles = K / TILE_K;

    auto issue = [&](int i, int b) {
        async_b128(pA0 + (long)i * TILE_K,     &As[b][sr * LDSS + sc]);
        async_b128(pA1 + (long)i * TILE_K,     &As[b][(sr + 32) * LDSS + sc]);
        async_b128(pB + (long)i * TILE_K * N,  &Bs[b][sr * LDSS + sc]);
    };

    issue(0, 0);
    for (int i = 0; i < ntiles; ++i) {
        const int b = i & 1;
        if (i + 1 < ntiles) {
            issue(i + 1, b ^ 1);
            __builtin_amdgcn_s_wait_asynccnt(3);  // tile i's 3 copies done
        } else {
            __builtin_amdgcn_s_wait_asynccnt(0);
        }
        __syncthreads();                          // tile i visible to all waves
        #pragma unroll
        for (int kk = 0; kk < TILE_K; kk += 4) {
            f32x2 a, bb;
            a.x  = As[b][ar * LDSS + kk + 2 * hl + 0];
            a.y  = As[b][ar * LDSS + kk + 2 * hl + 1];
            bb.x = Bs[b][(kk + 2 * hl + 0) * LDSS + bc];
            bb.y = Bs[b][(kk + 2 * hl + 1) * LDSS + bc];
            acc = __builtin_amdgcn_wmma_f32_16x16x4_f32(
                false, a, false, bb, (short)0, acc, false, false);
        }
        __syncthreads();   // all waves done with buf b before it is re-issued
    }
#else
    // fallback: synchronous staging through VGPRs, single buffer
    __shared__ float As[TILE_M * LDSS];
    __shared__ float Bs[TILE_K * LDSS];
    for (int k0 = 0; k0 < K; k0 += TILE_K) {
        #pragma unroll
        for (int t = 0; t < 2; ++t) {
            int idx = tid + t * 256;
            int row = idx >> 3;
            int c4  = (idx & 7) << 2;
            int gr  = m0 + row; gr = (gr < M) ? gr : (M - 1);
            const float4 v =
                *reinterpret_cast<const float4*>(A + (long)gr * K + k0 + c4);
            float* dst = &As[row * LDSS + c4];
            dst[0] = v.x; dst[1] = v.y; dst[2] = v.z; dst[3] = v.w;
            if (k0 + TILE_K < K)
                __builtin_prefetch(A + (long)gr * K + k0 + TILE_K + c4, 0, 1);
        }
        {
            int row = tid >> 3;
            int c4  = (tid & 7) << 2;
            const float4 v =
                *reinterpret_cast<const float4*>(B + (long)(k0 + row) * N + n0 + c4);
            float* dst = &Bs[row * LDSS + c4];
            dst[0] = v.x; dst[1] = v.y; dst[2] = v.z; dst[3] = v.w;
        }
        __syncthreads();
        #pragma unroll
        for (int kk = 0; kk < TILE_K; kk += 4) {
            f32x2 a, bb;
            a.x  = As[ar * LDSS + kk + 2 * hl + 0];
            a.y  = As[ar * LDSS + kk + 2 * hl + 1];
            bb.x = Bs[(kk + 2 * hl + 0) * LDSS + bc];
            bb.y = Bs[(kk + 2 * hl + 1) * LDSS + bc];
            acc = __builtin_amdgcn_wmma_f32_16x16x4_f32(
                false, a, false, bb, (short)0, acc, false, false);
        }
        __syncthreads();
    }
#endif

    // ---- store (C/D layout: VGPR v -> M = v (lanes0-15) / v+8 (lanes16-31)) ----
    const int tm = m0 + wr * 16;
    const int tn = n0 + wc * 16;
    #pragma unroll
    for (int r = 0; r < 8; ++r) {
        int row = tm + r + hl * 8;
        int col = tn + l16;
        if (row < M) {
            float o = acc[r];
            if (R) o += R[(long)row * N + col];
            C[(long)row * N + col] = o;
        }
    }
}

// ---------------------------------------------------------------------------
// CG pair coupling: cat[s, M, catOff + k] = sum_ij a[s,i,k] b[s,j,k] cg[i,j,M]
// One thread per (sample, channel). cg (<=245 floats) staged in LDS,
// uniform-address broadcast reads.
// ---------------------------------------------------------------------------
template<int D1, int D2, int DL>
__global__ __launch_bounds__(256) void cg_pair_kernel(
    const float* __restrict__ a, int lda,
    const float* __restrict__ b, int ldb,
    const float* __restrict__ cg,
    float* __restrict__ cat, int catStride, int catOff,
    int n, int mk)
{
    __shared__ float scg[D1 * D2 * DL];
    for (int i = threadIdx.x; i < D1 * D2 * DL; i += blockDim.x) scg[i] = cg[i];
    __syncthreads();

    unsigned idx = blockIdx.x * blockDim.x + threadIdx.x;
    if (idx >= (unsigned)(n * mk)) return;
    int k = (int)(idx % (unsigned)mk);
    int s = (int)(idx / (unsigned)mk);

    float av[D1], bv[D2];
    #pragma unroll
    for (int i = 0; i < D1; ++i) av[i] = a[(s * D1 + i) * lda + k];
    #pragma unroll
    for (int j = 0; j < D2; ++j) bv[j] = b[(s * D2 + j) * ldb + k];

    float ab[D1 * D2];
    #pragma unroll
    for (int i = 0; i < D1; ++i)
        #pragma unroll
        for (int j = 0; j < D2; ++j) ab[i * D2 + j] = av[i] * bv[j];

    #pragma unroll
    for (int Mo = 0; Mo < DL; ++Mo) {
        float s0 = 0.f;
        #pragma unroll
        for (int t = 0; t < D1 * D2; ++t) s0 += ab[t] * scg[t * DL + Mo];
        cat[(s * DL + Mo) * catStride + catOff + k] = s0;
    }
}

// ---------------------------------------------------------------------------
// Host side
// ---------------------------------------------------------------------------
static inline int imin(int a, int b) { return a < b ? a : b; }
static inline int imax(int a, int b) { return a > b ? a : b; }
static inline int iabs(int a)        { return a < 0 ? -a : a; }

static void launch_gemm(const float* A, const float* B, const float* R,
                        float* C, int M, int K, int N, hipStream_t s)
{
    dim3 grid((M + TILE_M - 1) / TILE_M, N / TILE_N);
    gemm_wmma_f32<<<grid, 256, 0, s>>>(A, B, R, C, M, K, N);
}

static void launch_cg(int l1, int l2, int L,
                      const float* f1, int K1, const float* f2, int K2,
                      const float* cg, float* cat, int catStride, int catOff,
                      int n, int mk, hipStream_t s)
{
    int d1 = 2 * l1 + 1, d2 = 2 * l2 + 1, dL = 2 * L + 1;
    int blocks = (n * mk + 255) / 256;
#define CGC(A_, B_, C_)                                                     \
    if (d1 == A_ && d2 == B_ && dL == C_) {                                 \
        cg_pair_kernel<A_, B_, C_><<<blocks, 256, 0, s>>>(                  \
            f1, K1, f2, K2, cg, cat, catStride, catOff, n, mk);             \
        return;                                                             \
    }
    CGC(1,1,1) CGC(3,3,1) CGC(5,5,1) CGC(7,7,1)
    CGC(1,3,3) CGC(3,1,3) CGC(3,5,3) CGC(5,3,3) CGC(5,7,3) CGC(7,5,3)
    CGC(1,5,5) CGC(3,3,5) CGC(3,7,5) CGC(5,1,5) CGC(5,5,5) CGC(7,3,5) CGC(7,7,5)
    CGC(1,7,7) CGC(3,5,7) CGC(5,3,7) CGC(5,7,7) CGC(7,1,7) CGC(7,5,7)
#undef CGC
}

extern "C" void kernel_launch(void* const* d_in, const int* in_sizes, int n_in,
                              void* d_out, int out_size, void* d_ws, size_t ws_size,
                              hipStream_t stream)
{
    static const int KL[4] = {256, 128, 64, 32};
    const int n = in_sizes[0] / 256;   // N_SAMPLES

    // ---- input pointer map (setup_inputs dict order / sorted nested keys) ----
    const float* feat[4];
    for (int l = 0; l < 4; ++l) feat[l] = (const float*)d_in[l];
    const float* mixw[3][4];
    for (int i = 0; i < 3; ++i)
        for (int l = 0; l < 4; ++l) mixw[i][l] = (const float*)d_in[4 + i * 4 + l];
    const float* linw[2][4];
    for (int it = 0; it < 2; ++it)
        for (int L = 0; L < 4; ++L) linw[it][L] = (const float*)d_in[16 + it * 4 + L];
    int cgIdx[4][4][4];
    {
        int p = 24;
        for (int l1 = 0; l1 < 4; ++l1)
            for (int l2 = 0; l2 < 4; ++l2)
                for (int L = iabs(l1 - l2); L <= imin(l1 + l2, 3); ++L)
                    cgIdx[l1][l2][L] = p++;
    }

    // ---- concat sizes per output L (even-parity pairs only) ----
    int sizeL[4] = {0, 0, 0, 0};
    for (int l1 = 0; l1 < 4; ++l1)
        for (int l2 = 0; l2 < 4; ++l2)
            for (int L = iabs(l1 - l2); L <= imin(l1 + l2, 3); ++L)
                if (((l1 + l2 + L) & 1) == 0) sizeL[L] += KL[imax(l1, l2)];
    // sizeL = {480, 448, 416, 256}

    // ---- workspace layout (floats) ----
    long blkF[4], prefF[4], T = 0;
    for (int l = 0; l < 4; ++l) {
        blkF[l] = (long)n * (2 * l + 1) * KL[l];
        prefF[l] = T;
        T += blkF[l];
    }
    float* ws = (float*)d_ws;
    float* mixed[3] = { ws, ws + T, ws + 2 * T };
    float* cur = ws + 3 * T;
    float* cat = ws + 4 * T;
    long catPref[4];
    {
        long acc = 0;
        for (int L = 0; L < 4; ++L) {
            catPref[L] = acc;
            acc += (long)n * (2 * L + 1) * sizeL[L];
        }
    }

    // ---- 1) mixers: mixed[i][l] = feat[l] x mixw[i][l] ----
    for (int i = 0; i < 3; ++i)
        for (int l = 0; l < 4; ++l)
            launch_gemm(feat[l], mixw[i][l], nullptr, mixed[i] + prefF[l],
                        n * (2 * l + 1), KL[l], KL[l], stream);

    // ---- 2) two CG iterations ----
    for (int it = 0; it < 2; ++it) {
        const float* f1 = (it == 0) ? (const float*)mixed[0] : (const float*)cur;
        const float* f2 = mixed[it + 1];

        int off[4] = {0, 0, 0, 0};
        for (int l1 = 0; l1 < 4; ++l1)
            for (int l2 = 0; l2 < 4; ++l2)
                for (int L = iabs(l1 - l2); L <= imin(l1 + l2, 3); ++L) {
                    if ((l1 + l2 + L) & 1) continue;
                    int mk = KL[imax(l1, l2)];
                    launch_cg(l1, l2, L,
                              f1 + prefF[l1], KL[l1],
                              f2 + prefF[l2], KL[l2],
                              (const float*)d_in[cgIdx[l1][l2][L]],
                              cat + catPref[L], sizeL[L], off[L],
                              n, mk, stream);
                    off[L] += mk;
                }

        for (int L = 0; L < 4; ++L) {
            const float* Rres = f1 + prefF[L];
            float* Cdst = (it == 0) ? (cur + prefF[L])
                                    : ((float*)d_out + prefF[L]);
            launch_gemm(cat + catPref[L], linw[it][L], Rres, Cdst,
                        n * (2 * L + 1), sizeL[L], KL[L], stream);
        }
    }
}